// WearPredictionGNN_9792525435128
// MI455X (gfx1250) — compile-verified
//
#include <hip/hip_runtime.h>

typedef __attribute__((ext_vector_type(2))) float v2f;
typedef __attribute__((ext_vector_type(8))) float v8f;

#define NN 100000
#define NE 1250000
#define HD 64
#define BN_EPS 1e-5f

// ---------------------------------------------------------------------------
// Zero-fill (graph-capture-safe scratch init)
// ---------------------------------------------------------------------------
__global__ void fill_zero_kernel(float* __restrict__ p, int n) {
  int i = blockIdx.x * blockDim.x + threadIdx.x;
  if (i < n) p[i] = 0.0f;
}

// ---------------------------------------------------------------------------
// WMMA f32 GEMM: out[M,HOUT] = epi( A[M,K]@W[K,HOUT] (+ A2[M,K]@W2) + bias )
// epi: optional bias -> optional BN(eval) -> optional ReLU -> optional +resid
//
// Block = (HOUT/16) waves. Wave w owns col tile w; each wave computes TWO
// 16x16 row tiles (rows blockIdx.x*32 .. +32). W fragments for the whole K
// range are preloaded into registers (K/2 floats per lane) and reused across
// both row tiles. All loops fully unrolled at compile time -> branch-free,
// clause-friendly codegen around v_wmma_f32_16x16x4_f32.
// Requires M % 32 == 0 (100000 = 3125 * 32).
// ---------------------------------------------------------------------------
template <int K, int HOUT, bool HAS2, bool BIAS, bool BN, bool RELU, bool RES>
__global__ void __launch_bounds__((HOUT / 16) * 32)
wmma_gemm_kernel(const float* __restrict__ A, const float* __restrict__ W,
                 const float* __restrict__ A2, const float* __restrict__ W2,
                 const float* __restrict__ bias,
                 const float* __restrict__ gamma, const float* __restrict__ beta,
                 const float* __restrict__ resid,
                 float* __restrict__ out) {
  constexpr int NS = K / 4;            // WMMA K-steps
  const int lane = threadIdx.x & 31;
  const int wave = threadIdx.x >> 5;   // column tile
  const int row0 = blockIdx.x * 32;
  const int col0 = wave * 16;
  const int r    = lane & 15;
  const int half = lane >> 4;          // 0: K=k,k+1   1: K=k+2,k+3

  // Preload B fragments (W slice for this col tile) into registers.
  v2f bf[NS];
#pragma unroll
  for (int s = 0; s < NS; ++s) {
    const int kk = 4 * s + half * 2;
    bf[s].x = W[kk * HOUT + col0 + r];
    bf[s].y = W[(kk + 1) * HOUT + col0 + r];
  }
  v2f bf2[HAS2 ? NS : 1];
  if (HAS2) {
#pragma unroll
    for (int s = 0; s < NS; ++s) {
      const int kk = 4 * s + half * 2;
      bf2[s].x = W2[kk * HOUT + col0 + r];
      bf2[s].y = W2[(kk + 1) * HOUT + col0 + r];
    }
  }

  v8f acc[2] = {v8f{}, v8f{}};
#pragma unroll
  for (int t = 0; t < 2; ++t) {
    const float* arow = A + (size_t)(row0 + t * 16 + r) * K;
#pragma unroll
    for (int s = 0; s < NS; ++s) {
      const int kk = 4 * s + half * 2;
      v2f a;
      a.x = arow[kk];
      a.y = arow[kk + 1];
      acc[t] = __builtin_amdgcn_wmma_f32_16x16x4_f32(false, a, false, bf[s],
                                                     (short)0, acc[t], false, false);
    }
    if (HAS2) {
      const float* arow2 = A2 + (size_t)(row0 + t * 16 + r) * K;
#pragma unroll
      for (int s = 0; s < NS; ++s) {
        const int kk = 4 * s + half * 2;
        v2f a;
        a.x = arow2[kk];
        a.y = arow2[kk + 1];
        acc[t] = __builtin_amdgcn_wmma_f32_16x16x4_f32(false, a, false, bf2[s],
                                                       (short)0, acc[t], false, false);
      }
    }
  }

  // Epilogue. C/D layout: vgpr i -> row i (lanes 0-15) / row i+8 (lanes 16-31),
  // col = lane & 15.
  const float rs = rsqrtf(1.0f + BN_EPS);
  const int n = col0 + r;
#pragma unroll
  for (int t = 0; t < 2; ++t) {
#pragma unroll
    for (int i = 0; i < 8; ++i) {
      const int m = row0 + t * 16 + i + half * 8;
      float v = acc[t][i];
      if (BIAS) v += bias[n];
      if (BN)   v = v * (gamma[n] * rs) + beta[n];
      if (RELU) v = fmaxf(v, 0.0f);
      if (RES)  v += resid[(size_t)m * HOUT + n];
      out[(size_t)m * HOUT + n] = v;
    }
  }
}

// ---------------------------------------------------------------------------
// Per-node attention partial dots: s1[n]=hlin[n]·Wa[0:64], s2[n]=hlin[n]·Wa[64:128]
// ---------------------------------------------------------------------------
__global__ void att_scores_kernel(const float* __restrict__ hlin,
                                  const float* __restrict__ Wa,
                                  float* __restrict__ s12) {
  int i = blockIdx.x * blockDim.x + threadIdx.x;
  if (i >= NN) return;
  const float* h = hlin + (size_t)i * HD;
  float a = 0.0f, b = 0.0f;
#pragma unroll 8
  for (int j = 0; j < HD; ++j) {
    a += h[j] * Wa[j];
    b += h[j] * Wa[HD + j];
  }
  s12[i]      = a;
  s12[NN + i] = b;
}

// ---------------------------------------------------------------------------
// Edge-attention scatter: agg[dst] += hlin[src] * sigmoid(leaky_relu(score))
// 16 lanes per edge, float4 per lane.
// ---------------------------------------------------------------------------
__global__ void edge_att_scatter_kernel(const float* __restrict__ hlin,
                                        const float* __restrict__ s12,
                                        const float* __restrict__ eattr,
                                        const int* __restrict__ src,
                                        const int* __restrict__ dst,
                                        const float* __restrict__ Wa,
                                        const float* __restrict__ ba,
                                        float* __restrict__ agg) {
  long long t = (long long)blockIdx.x * blockDim.x + threadIdx.x;
  int e = (int)(t >> 4);
  int l = (int)(t & 15);
  if (e >= NE) return;
  const int s = src[e], d = dst[e];
  float z = s12[d] + s12[NN + s]
          + eattr[e * 3 + 0] * Wa[128]
          + eattr[e * 3 + 1] * Wa[129]
          + eattr[e * 3 + 2] * Wa[130] + ba[0];
  z = (z > 0.0f) ? z : 0.01f * z;                 // leaky_relu(0.01)
  const float a = 1.0f / (1.0f + __expf(-z));     // sigmoid
  const float4 v = *(const float4*)(hlin + (size_t)s * HD + l * 4);
  float* o = agg + (size_t)d * HD + l * 4;
  atomicAdd(o + 0, v.x * a);
  atomicAdd(o + 1, v.y * a);
  atomicAdd(o + 2, v.z * a);
  atomicAdd(o + 3, v.w * a);
}

// out = relu(bn(in)) + resid      (per-column gamma/beta, 64 cols)
__global__ void bn_relu_res_kernel(const float* __restrict__ in,
                                   const float* __restrict__ gamma,
                                   const float* __restrict__ beta,
                                   const float* __restrict__ resid,
                                   float* __restrict__ out, int total) {
  int i = blockIdx.x * blockDim.x + threadIdx.x;
  if (i >= total) return;
  const int c = i & (HD - 1);
  const float rs = rsqrtf(1.0f + BN_EPS);
  float v = in[i] * (gamma[c] * rs) + beta[c];
  v = fmaxf(v, 0.0f);
  out[i] = v + resid[i];
}

// SAGE scatter: agg[dst] += h[src]; cnt[dst] += 1
__global__ void sage_scatter_kernel(const float* __restrict__ h,
                                    const int* __restrict__ src,
                                    const int* __restrict__ dst,
                                    float* __restrict__ agg,
                                    float* __restrict__ cnt) {
  long long t = (long long)blockIdx.x * blockDim.x + threadIdx.x;
  int e = (int)(t >> 4);
  int l = (int)(t & 15);
  if (e >= NE) return;
  const int s = src[e], d = dst[e];
  const float4 v = *(const float4*)(h + (size_t)s * HD + l * 4);
  float* o = agg + (size_t)d * HD + l * 4;
  atomicAdd(o + 0, v.x);
  atomicAdd(o + 1, v.y);
  atomicAdd(o + 2, v.z);
  atomicAdd(o + 3, v.w);
  if (l == 0) atomicAdd(cnt + d, 1.0f);
}

// agg[n,:] /= max(cnt[n],1)
__global__ void mean_kernel(float* __restrict__ agg, const float* __restrict__ cnt,
                            int total) {
  int i = blockIdx.x * blockDim.x + threadIdx.x;
  if (i >= total) return;
  agg[i] = agg[i] / fmaxf(cnt[i >> 6], 1.0f);
}

// GCN scatter: agg[dst] += hw[src] * rsqrt(deg[src]) * rsqrt(deg[dst]),  deg = cnt+1
__global__ void gcn_scatter_kernel(const float* __restrict__ hw,
                                   const int* __restrict__ src,
                                   const int* __restrict__ dst,
                                   const float* __restrict__ cnt,
                                   float* __restrict__ agg) {
  long long t = (long long)blockIdx.x * blockDim.x + threadIdx.x;
  int e = (int)(t >> 4);
  int l = (int)(t & 15);
  if (e >= NE) return;
  const int s = src[e], d = dst[e];
  const float norm = rsqrtf(cnt[s] + 1.0f) * rsqrtf(cnt[d] + 1.0f);
  const float4 v = *(const float4*)(hw + (size_t)s * HD + l * 4);
  float* o = agg + (size_t)d * HD + l * 4;
  atomicAdd(o + 0, v.x * norm);
  atomicAdd(o + 1, v.y * norm);
  atomicAdd(o + 2, v.z * norm);
  atomicAdd(o + 3, v.w * norm);
}

// out = relu(bn(agg + hw/deg + bias)) + resid    (self-loop term hw/deg folded in)
__global__ void gcn_finish_kernel(const float* __restrict__ agg,
                                  const float* __restrict__ hw,
                                  const float* __restrict__ cnt,
                                  const float* __restrict__ bias,
                                  const float* __restrict__ gamma,
                                  const float* __restrict__ beta,
                                  const float* __restrict__ resid,
                                  float* __restrict__ out, int total) {
  int i = blockIdx.x * blockDim.x + threadIdx.x;
  if (i >= total) return;
  const int n = i >> 6;
  const int c = i & (HD - 1);
  const float rs = rsqrtf(1.0f + BN_EPS);
  float v = agg[i] + hw[i] / (cnt[n] + 1.0f) + bias[c];
  v = v * (gamma[c] * rs) + beta[c];
  v = fmaxf(v, 0.0f);
  out[i] = v + resid[i];
}

// out[n] = h[n,0:32]·W_r3 + b_r3
__global__ void final_kernel(const float* __restrict__ h,
                             const float* __restrict__ W,
                             const float* __restrict__ b,
                             float* __restrict__ out) {
  int i = blockIdx.x * blockDim.x + threadIdx.x;
  if (i >= NN) return;
  float acc = b[0];
#pragma unroll
  for (int j = 0; j < 32; ++j) acc += h[i * 32 + j] * W[j];
  out[i] = acc;
}

// ---------------------------------------------------------------------------
extern "C" void kernel_launch(void* const* d_in, const int* in_sizes, int n_in,
                              void* d_out, int out_size, void* d_ws, size_t ws_size,
                              hipStream_t stream) {
  const float* x       = (const float*)d_in[0];
  const float* eattr   = (const float*)d_in[1];
  const int*   eidx    = (const int*)  d_in[2];
  const float* W_node  = (const float*)d_in[3];
  const float* b_node  = (const float*)d_in[4];
  // d_in[5], d_in[6]: W_edge/b_edge -> result unused in reference
  const float* W_lin0  = (const float*)d_in[7];
  const float* b_lin0  = (const float*)d_in[8];
  const float* W_att0  = (const float*)d_in[9];
  const float* b_att0  = (const float*)d_in[10];
  const float* W_sage_l= (const float*)d_in[11];
  const float* b_sage_l= (const float*)d_in[12];
  const float* W_sage_r= (const float*)d_in[13];
  const float* W_gcn   = (const float*)d_in[14];
  const float* b_gcn   = (const float*)d_in[15];
  const float* bn_g    = (const float*)d_in[16];   // [3,64]
  const float* bn_b    = (const float*)d_in[17];
  const float* W_r1    = (const float*)d_in[18];
  const float* b_r1    = (const float*)d_in[19];
  const float* W_r2    = (const float*)d_in[20];
  const float* b_r2    = (const float*)d_in[21];
  const float* W_r3    = (const float*)d_in[22];
  const float* b_r3    = (const float*)d_in[23];
  float* out = (float*)d_out;

  const int* src = eidx;
  const int* dst = eidx + NE;

  const size_t N64 = (size_t)NN * HD;
  float* A   = (float*)d_ws;          // [N,64]
  float* B   = A + N64;               // [N,64]
  float* C   = B + N64;               // [N,64] scatter accumulator
  float* D   = C + N64;               // [N,64]
  float* s12 = D + N64;               // [2N]
  float* cnt = s12 + 2 * NN;          // [N]

  const dim3 gM(NN / 32);             // 3125 blocks, 2 row tiles per wave
  const dim3 blk128(128);             // HOUT=64 -> 4 waves (one per col tile)
  const dim3 blk64(64);               // HOUT=32 -> 2 waves
  const int TB = 256;
  const int nbN   = (NN + TB - 1) / TB;
  const int nbN64 = (int)((N64 + TB - 1) / TB);
  const int nbE16 = (int)(((long long)NE * 16 + TB - 1) / TB);

  // h0 = x @ W_node + b_node                                  -> A
  wmma_gemm_kernel<16, 64, false, true, false, false, false>
      <<<gM, blk128, 0, stream>>>(x, W_node, nullptr, nullptr, b_node,
                                  nullptr, nullptr, nullptr, A);
  // hlin = h0 @ W_lin0 + b_lin0                               -> B
  wmma_gemm_kernel<64, 64, false, true, false, false, false>
      <<<gM, blk128, 0, stream>>>(A, W_lin0, nullptr, nullptr, b_lin0,
                                  nullptr, nullptr, nullptr, B);
  // per-node attention partials
  att_scores_kernel<<<nbN, TB, 0, stream>>>(B, W_att0, s12);
  // attention scatter into C
  fill_zero_kernel<<<nbN64, TB, 0, stream>>>(C, (int)N64);
  edge_att_scatter_kernel<<<nbE16, TB, 0, stream>>>(B, s12, eattr, src, dst,
                                                    W_att0, b_att0, C);
  // h1 = relu(bn0(C)) + h0                                    -> D
  bn_relu_res_kernel<<<nbN64, TB, 0, stream>>>(C, bn_g + 0, bn_b + 0, A, D, (int)N64);

  // SAGE: mean aggregation
  fill_zero_kernel<<<nbN64, TB, 0, stream>>>(C, (int)N64);
  fill_zero_kernel<<<nbN, TB, 0, stream>>>(cnt, NN);
  sage_scatter_kernel<<<nbE16, TB, 0, stream>>>(D, src, dst, C, cnt);
  mean_kernel<<<nbN64, TB, 0, stream>>>(C, cnt, (int)N64);
  // h2 = relu(bn1(mean@Wl + h1@Wr + bl)) + h1                 -> A
  wmma_gemm_kernel<64, 64, true, true, true, true, true>
      <<<gM, blk128, 0, stream>>>(C, W_sage_l, D, W_sage_r, b_sage_l,
                                  bn_g + 64, bn_b + 64, D, A);

  // GCN: hw = h2 @ W_gcn                                      -> B
  wmma_gemm_kernel<64, 64, false, false, false, false, false>
      <<<gM, blk128, 0, stream>>>(A, W_gcn, nullptr, nullptr, nullptr,
                                  nullptr, nullptr, nullptr, B);
  fill_zero_kernel<<<nbN64, TB, 0, stream>>>(C, (int)N64);
  gcn_scatter_kernel<<<nbE16, TB, 0, stream>>>(B, src, dst, cnt, C);
  // h3 = relu(bn2(C + hw/deg + b_gcn)) + h2                   -> D
  gcn_finish_kernel<<<nbN64, TB, 0, stream>>>(C, B, cnt, b_gcn, bn_g + 128, bn_b + 128,
                                              A, D, (int)N64);

  // regressor
  wmma_gemm_kernel<64, 64, false, true, false, true, false>
      <<<gM, blk128, 0, stream>>>(D, W_r1, nullptr, nullptr, b_r1,
                                  nullptr, nullptr, nullptr, B);
  wmma_gemm_kernel<64, 32, false, true, false, true, false>
      <<<gM, blk64, 0, stream>>>(B, W_r2, nullptr, nullptr, b_r2,
                                 nullptr, nullptr, nullptr, C);
  final_kernel<<<nbN, TB, 0, stream>>>(C, W_r3, b_r3, out);
}